// SOM_64037962383633
// MI455X (gfx1250) — compile-verified
//
#include <hip/hip_runtime.h>
#include <math.h>

// ---------------------------------------------------------------------------
// SOM update for MI455X (gfx1250, wave32).
//   B=32 batches, G=64 (grid 64x64 = 4096 units), C=768 features.
//   Bandwidth-bound (~38MB traffic); fp32 WMMA (V_WMMA_F32_16X16X4_F32) for
//   both GEMMs: dot = units @ x^T  and  acc = W @ x.
//   unorm2 is fused into the dot GEMM (units is read exactly once there).
// ---------------------------------------------------------------------------

#define SOM_G   64
#define SOM_GG  4096      // G*G
#define SOM_B   32
#define SOM_C   768
#define SOM_LR  0.01f

typedef float v2f __attribute__((ext_vector_type(2)));
typedef float v8f __attribute__((ext_vector_type(8)));

static __device__ __forceinline__ v8f wmma_f32_16x16x4(v2f a, v2f b, v8f c) {
    // 8-arg pattern: (neg_a, A, neg_b, B, c_mod, C, reuse_a, reuse_b)
    return __builtin_amdgcn_wmma_f32_16x16x4_f32(false, a, false, b,
                                                 (short)0, c, false, false);
}

// ---------------------------------------------------------------------------
// Kernel 1: DOT[g,b] = sum_c units[g,c] * x[b,c]   (GEMM 4096x32x768, fp32)
//           + fused unorm2[g] = sum_c units[g,c]^2 from the staged LDS tile.
// Block = 256 threads = 8 waves; block tile = 64(g) x 32(b).
// Waves laid out 4(g-subtiles) x 2(b-subtiles) of 16x16 each.
// K staged in LDS 32 at a time (padded stride 36 floats).
// ---------------------------------------------------------------------------
__global__ void som_dot_kernel(const float* __restrict__ units,
                               const float* __restrict__ x,
                               float* __restrict__ DOT,
                               float* __restrict__ unorm2) {
    __shared__ float At[64][36];   // units tile [g][k]
    __shared__ float Bt[32][36];   // x tile     [b][k]

    const int tid   = threadIdx.x;
    const int wave  = tid >> 5;
    const int lane  = tid & 31;
    const int grp   = lane >> 4;        // lane half selects K pair {0,1}/{2,3}
    const int ln15  = lane & 15;
    const int wgs   = wave >> 1;        // g-subtile 0..3
    const int wbs   = wave & 1;         // b-subtile 0..1
    const int gbase = blockIdx.x * 64;

    // fused unorm2: 4 threads per g-row, 8 staged elements each
    const int nrow = tid >> 2;          // 0..63
    const int ncol = (tid & 3) * 8;     // 0,8,16,24
    float usq = 0.0f;

    v8f acc = {};

    for (int kb = 0; kb < SOM_C; kb += 32) {
        // ---- stage units 64x32 (2 passes, b128, fully coalesced) ----
#pragma unroll
        for (int p = 0; p < 2; ++p) {
            int idx  = tid + p * 256;
            int row  = idx >> 3;
            int col4 = (idx & 7) * 4;
            float4 v = *(const float4*)&units[(size_t)(gbase + row) * SOM_C + kb + col4];
            *(float4*)&At[row][col4] = v;
        }
        // ---- stage x 32x32 (1 pass) ----
        {
            int row  = tid >> 3;
            int col4 = (tid & 7) * 4;
            float4 v = *(const float4*)&x[(size_t)row * SOM_C + kb + col4];
            *(float4*)&Bt[row][col4] = v;
        }
        __syncthreads();

        // ---- fused ||u||^2 partial from the staged tile ----
#pragma unroll
        for (int j = 0; j < 8; ++j) {
            float u = At[nrow][ncol + j];
            usq += u * u;
        }

        // ---- 8 WMMAs, K=4 each ----
#pragma unroll
        for (int i = 0; i < 8; ++i) {
            const int kk = 4 * i + 2 * grp;
            v2f a = *(const v2f*)&At[wgs * 16 + ln15][kk];  // A[m][k..k+1]
            v2f b = *(const v2f*)&Bt[wbs * 16 + ln15][kk];  // B[k..k+1][n]
            acc = wmma_f32_16x16x4(a, b, acc);
        }
        __syncthreads();
    }

    // reduce the 4 partials per row (lanes tid&3 are wave-adjacent)
    usq += __shfl_xor(usq, 1, 32);
    usq += __shfl_xor(usq, 2, 32);
    if ((tid & 3) == 0) unorm2[gbase + nrow] = usq;

    // C/D layout: VGPR r -> M = r + 8*grp ; lane&15 -> N
#pragma unroll
    for (int r = 0; r < 8; ++r) {
        int g = gbase + wgs * 16 + r + 8 * grp;
        int b = wbs * 16 + ln15;
        DOT[(size_t)g * SOM_B + b] = acc[r];
    }
}

// ---------------------------------------------------------------------------
// Kernel 2: argmin per batch. dist_rel = unorm2[g] - 2*dot (xnorm dropped).
// 32 blocks (one per batch) x 256 threads; strided scan keeps lowest index
// per thread, all combines are lower-index-wins on ties (first occurrence).
// ---------------------------------------------------------------------------
__global__ void som_argmin_kernel(const float* __restrict__ DOT,
                                  const float* __restrict__ unorm2,
                                  int* __restrict__ bmu) {
    __shared__ float sval[8];
    __shared__ int   sidx[8];

    const int b    = blockIdx.x;
    const int tid  = threadIdx.x;
    const int wave = tid >> 5;
    const int lane = tid & 31;

    float best = 3.4e38f;
    int   bidx = SOM_GG;
    for (int g = tid; g < SOM_GG; g += 256) {       // increasing g per thread
        float d = unorm2[g] - 2.0f * DOT[(size_t)g * SOM_B + b];
        if (d < best) { best = d; bidx = g; }
    }
#pragma unroll
    for (int off = 16; off; off >>= 1) {
        float ov = __shfl_xor(best, off, 32);
        int   oi = __shfl_xor(bidx, off, 32);
        if (ov < best || (ov == best && oi < bidx)) { best = ov; bidx = oi; }
    }
    if (lane == 0) { sval[wave] = best; sidx[wave] = bidx; }
    __syncthreads();
    if (tid == 0) {
        best = sval[0]; bidx = sidx[0];
#pragma unroll
        for (int w = 1; w < 8; ++w) {
            float ov = sval[w]; int oi = sidx[w];
            if (ov < best || (ov == best && oi < bidx)) { best = ov; bidx = oi; }
        }
        bmu[b] = bidx;
    }
}

// ---------------------------------------------------------------------------
// Kernel 3: neighborhood weights. w[b,g] = (temp/B)*gaussian (cutoff applied),
// S[g] = sum_b w[b,g].  Schedule scalars computed on-device from t.
// ---------------------------------------------------------------------------
__global__ void som_weights_kernel(const int* __restrict__ bmu,
                                   const int* __restrict__ tptr,
                                   float* __restrict__ W,
                                   float* __restrict__ S) {
    const int g = blockIdx.x * 256 + threadIdx.x;   // exactly 4096 threads
    const float tf = (float)(*tptr);
    const float deno  = logf(64.0f) / 9000.0f;       // 1/(0.9*MAX_T/log G)
    const float ksize = 64.0f * expf(-tf * deno);
    const float sigma = 0.3f * ((ksize - 1.0f) * 0.5f - 1.0f) + 0.8f;
    const float temperature = expf(-(tf * 2.0f) / 10000.0f);
    const float inv2s2 = 1.0f / (2.0f * sigma * sigma);
    const float gy = (float)(g >> 6);
    const float gx = (float)(g & 63);

    float s = 0.0f;
#pragma unroll
    for (int b = 0; b < SOM_B; ++b) {
        int   m  = bmu[b];
        float by = (float)(m >> 6);
        float bx = (float)(m & 63);
        float pd = (gy - by) * (gy - by) + (gx - bx) * (gx - bx);
        float ga = expf(-pd * inv2s2);
        ga = (ga < 0.001f) ? 0.0f : ga;
        float w = temperature * ga * (1.0f / (float)SOM_B);
        W[(size_t)g * SOM_B + b] = w;
        s += w;
    }
    S[g] = s;
}

// ---------------------------------------------------------------------------
// Kernel 4: acc[g,c] = sum_b W[g,b]*x[b,c]  (GEMM 4096x768x32, fp32 WMMA)
// then out[g,c] = units[g,c]*(1 - LR*S[g]) + LR*acc[g,c].
// Block = 256 threads = 8 waves; tile = 32(g) x 64(c); waves 2x4.
// ---------------------------------------------------------------------------
__global__ void som_update_kernel(const float* __restrict__ units,
                                  const float* __restrict__ x,
                                  const float* __restrict__ W,
                                  const float* __restrict__ S,
                                  float* __restrict__ out) {
    __shared__ float Wt[32][36];    // W tile  [g][b]
    __shared__ float Xt[64][36];    // x^T tile [c][b]

    const int tid   = threadIdx.x;
    const int wave  = tid >> 5;
    const int lane  = tid & 31;
    const int grp   = lane >> 4;
    const int ln15  = lane & 15;
    const int wgs   = wave >> 2;        // g-subtile 0..1
    const int wcs   = wave & 3;         // c-subtile 0..3
    const int gbase = blockIdx.y * 32;
    const int cbase = blockIdx.x * 64;

    // ---- stage W 32x32 ----
    {
        int row  = tid >> 3;
        int col4 = (tid & 7) * 4;
        float4 v = *(const float4*)&W[(size_t)(gbase + row) * SOM_B + col4];
        *(float4*)&Wt[row][col4] = v;
    }
    // ---- stage x^T 64(c) x 32(b): coalesced b128 reads, scattered LDS writes
#pragma unroll
    for (int p = 0; p < 2; ++p) {
        int idx  = tid + p * 256;
        int brow = idx >> 4;            // batch 0..31
        int col4 = (idx & 15) * 4;      // c offset 0..60
        float4 v = *(const float4*)&x[(size_t)brow * SOM_C + cbase + col4];
        Xt[col4 + 0][brow] = v.x;
        Xt[col4 + 1][brow] = v.y;
        Xt[col4 + 2][brow] = v.z;
        Xt[col4 + 3][brow] = v.w;
    }
    __syncthreads();

    v8f acc = {};
#pragma unroll
    for (int i = 0; i < 8; ++i) {       // K = 32 = 8 x 4
        const int kk = 4 * i + 2 * grp;
        v2f a = *(const v2f*)&Wt[wgs * 16 + ln15][kk];
        v2f b = *(const v2f*)&Xt[wcs * 16 + ln15][kk];
        acc = wmma_f32_16x16x4(a, b, acc);
    }

    // ---- fused epilogue: out = units*(1-LR*S) + LR*acc ----
#pragma unroll
    for (int r = 0; r < 8; ++r) {
        int g = gbase + wgs * 16 + r + 8 * grp;
        int c = cbase + wcs * 16 + ln15;
        size_t o = (size_t)g * SOM_C + c;
        float u = units[o];
        float s = S[g];
        out[o] = u * (1.0f - SOM_LR * s) + SOM_LR * acc[r];
    }
}

// ---------------------------------------------------------------------------
extern "C" void kernel_launch(void* const* d_in, const int* in_sizes, int n_in,
                              void* d_out, int out_size, void* d_ws, size_t ws_size,
                              hipStream_t stream) {
    (void)in_sizes; (void)n_in; (void)out_size; (void)ws_size;

    const float* x     = (const float*)d_in[0];   // [32,3,16,16] -> [32,768]
    const float* units = (const float*)d_in[1];   // [64,64,768]  -> [4096,768]
    const int*   tptr  = (const int*)d_in[2];     // scalar t
    float* out = (float*)d_out;

    // workspace layout (~1.03 MB)
    float* DOT    = (float*)d_ws;                 // 4096*32
    float* unorm2 = DOT + (size_t)SOM_GG * SOM_B; // 4096
    int*   bmu    = (int*)(unorm2 + SOM_GG);      // 32
    float* W      = (float*)(bmu + SOM_B);        // 4096*32
    float* S      = W + (size_t)SOM_GG * SOM_B;   // 4096

    som_dot_kernel<<<SOM_GG / 64, 256, 0, stream>>>(units, x, DOT, unorm2);
    som_argmin_kernel<<<SOM_B, 256, 0, stream>>>(DOT, unorm2, bmu);
    som_weights_kernel<<<SOM_GG / 256, 256, 0, stream>>>(bmu, tptr, W, S);
    som_update_kernel<<<dim3(SOM_C / 64, SOM_GG / 32), 256, 0, stream>>>(
        units, x, W, S, out);
}